// Decoder_24919400252011
// MI455X (gfx1250) — compile-verified
//
#include <hip/hip_runtime.h>
#include <hip/hip_bf16.h>

typedef __attribute__((ext_vector_type(2))) float v2f;
typedef __attribute__((ext_vector_type(8))) float v8f;

#define N_NODES 8192
#define N_CODES 4096
#define DIM     256
#define EPS     1e-8f
#define NCHUNKS 8          // column chunks for cosine partial argmax (4096/512)

// ---------------------------------------------------------------------------
// Kernel 1: adj = sigmoid(z @ z^T), f32 WMMA 16x16x4.
// Block = 256 threads (8 waves). Block tile: 256(M) x 128(N).
// Wave tile: 64x64 = 4x4 WMMA tiles (128 accumulator VGPRs).
// A-frag lane map: row = m0 + lane%16, k = kbase + 2*(lane/16) -> float2.
// B = z^T so B-frag uses the identical gather with column index.
// ---------------------------------------------------------------------------
__global__ __launch_bounds__(256) void adj_sigmoid_kernel(
    const float* __restrict__ z, float* __restrict__ out) {
  const int lane  = threadIdx.x & 31;
  const int wave  = threadIdx.x >> 5;
  const int wm    = wave & 3;            // 4 waves along M
  const int wn    = wave >> 2;           // 2 waves along N
  const int m0    = blockIdx.y * 256 + wm * 64;
  const int n0    = blockIdx.x * 128 + wn * 64;
  const int lrow  = lane & 15;
  const int khalf = (lane >> 4) << 1;    // 0 or 2

  v8f acc[4][4] = {};

  const float* Abase = z + (size_t)(m0 + lrow) * DIM + khalf;
  const float* Bbase = z + (size_t)(n0 + lrow) * DIM + khalf;

#pragma unroll 2
  for (int k = 0; k < DIM; k += 4) {
    v2f a[4], b[4];
#pragma unroll
    for (int t = 0; t < 4; ++t) {
      a[t] = *(const v2f*)(Abase + (size_t)(t * 16) * DIM + k);
      b[t] = *(const v2f*)(Bbase + (size_t)(t * 16) * DIM + k);
    }
#pragma unroll
    for (int ti = 0; ti < 4; ++ti)
#pragma unroll
      for (int tj = 0; tj < 4; ++tj)
        acc[ti][tj] = __builtin_amdgcn_wmma_f32_16x16x4_f32(
            false, a[ti], false, b[tj], (short)0, acc[ti][tj], false, false);
  }

  // C/D layout: lane L, vgpr v -> M = v + 8*(L/16), N = L%16
  const int mhalf = (lane >> 4) << 3;    // 0 or 8
#pragma unroll
  for (int ti = 0; ti < 4; ++ti) {
#pragma unroll
    for (int tj = 0; tj < 4; ++tj) {
      const size_t col = (size_t)(n0 + tj * 16 + lrow);
#pragma unroll
      for (int v = 0; v < 8; ++v) {
        const size_t row = (size_t)(m0 + ti * 16 + mhalf + v);
        const float x = acc[ti][tj][v];
        out[row * N_NODES + col] = 1.0f / (1.0f + __expf(-x));
      }
    }
  }
}

// ---------------------------------------------------------------------------
// Kernel 2: rn[j] = 1 / max(||e_j||, eps)  (norms of codebook rows)
// ---------------------------------------------------------------------------
__global__ void enorm_kernel(const float* __restrict__ E, float* __restrict__ rn) {
  const int row = blockIdx.x * blockDim.x + threadIdx.x;
  if (row < N_CODES) {
    const float4* p = (const float4*)(E + (size_t)row * DIM);
    float s = 0.0f;
#pragma unroll 8
    for (int i = 0; i < DIM / 4; ++i) {
      const float4 q = p[i];
      s += q.x * q.x + q.y * q.y + q.z * q.z + q.w * q.w;
    }
    rn[row] = 1.0f / fmaxf(sqrtf(s), EPS);
  }
}

// ---------------------------------------------------------------------------
// Kernel 3: partial argmax_j (z_i . e_j) * rn[j] over a 512-column chunk.
// Row normalization of z is argmax-invariant -> skipped.
// Block = 256 threads (8 waves), 16 rows per wave, grid (NCHUNKS, 64).
// Running (val,idx) kept per (lane,vgpr) slot — same output row across
// tiles/chunk-iters — then shfl_xor reduced across the 16-lane half.
// ---------------------------------------------------------------------------
__global__ __launch_bounds__(256) void cos_partial_kernel(
    const float* __restrict__ z, const float* __restrict__ E,
    const float* __restrict__ rn, float* __restrict__ pval,
    int* __restrict__ pidx) {
  const int lane  = threadIdx.x & 31;
  const int wave  = threadIdx.x >> 5;
  const int lrow  = lane & 15;
  const int khalf = (lane >> 4) << 1;
  const int m0    = blockIdx.y * 128 + wave * 16;
  const int chunk = blockIdx.x;
  const int n0    = chunk * (N_CODES / NCHUNKS);   // 512-wide chunk

  float bestv[8];
  int   besti[8];
#pragma unroll
  for (int v = 0; v < 8; ++v) { bestv[v] = -3.4e38f; besti[v] = 0; }

  const float* Abase = z + (size_t)(m0 + lrow) * DIM + khalf;

  for (int nc = 0; nc < N_CODES / NCHUNKS; nc += 64) {
    const float* Bbase = E + (size_t)(n0 + nc + lrow) * DIM + khalf;
    v8f acc[4] = {};
#pragma unroll 2
    for (int k = 0; k < DIM; k += 4) {
      const v2f a = *(const v2f*)(Abase + k);
#pragma unroll
      for (int tj = 0; tj < 4; ++tj) {
        const v2f b = *(const v2f*)(Bbase + (size_t)(tj * 16) * DIM + k);
        acc[tj] = __builtin_amdgcn_wmma_f32_16x16x4_f32(
            false, a, false, b, (short)0, acc[tj], false, false);
      }
    }
#pragma unroll
    for (int tj = 0; tj < 4; ++tj) {
      const int n = n0 + nc + tj * 16 + lrow;
      const float r = rn[n];
#pragma unroll
      for (int v = 0; v < 8; ++v) {
        const float val = acc[tj][v] * r;
        if (val > bestv[v] || (val == bestv[v] && n < besti[v])) {
          bestv[v] = val; besti[v] = n;
        }
      }
    }
  }

  // reduce across the 16 lanes of each half (same row per (lane-half, v))
#pragma unroll
  for (int mask = 1; mask < 16; mask <<= 1) {
#pragma unroll
    for (int v = 0; v < 8; ++v) {
      const float ov = __shfl_xor(bestv[v], mask, 32);
      const int   oi = __shfl_xor(besti[v], mask, 32);
      if (ov > bestv[v] || (ov == bestv[v] && oi < besti[v])) {
        bestv[v] = ov; besti[v] = oi;
      }
    }
  }

  if (lrow == 0) {
    const int rbase = m0 + ((lane >> 4) << 3);
#pragma unroll
    for (int v = 0; v < 8; ++v) {
      const size_t row = (size_t)(rbase + v);
      pval[(size_t)chunk * N_NODES + row] = bestv[v];
      pidx[(size_t)chunk * N_NODES + row] = besti[v];
    }
  }
}

// ---------------------------------------------------------------------------
// Kernel 4: finalize argmax over the NCHUNKS partials, write index as float.
// ---------------------------------------------------------------------------
__global__ void argmax_finalize_kernel(const float* __restrict__ pval,
                                       const int* __restrict__ pidx,
                                       float* __restrict__ outIdx) {
  const int row = blockIdx.x * blockDim.x + threadIdx.x;
  if (row < N_NODES) {
    float bv = -3.4e38f;
    int   bi = 0;
#pragma unroll
    for (int c = 0; c < NCHUNKS; ++c) {
      const float v = pval[(size_t)c * N_NODES + row];
      const int   i = pidx[(size_t)c * N_NODES + row];
      if (v > bv || (v == bv && i < bi)) { bv = v; bi = i; }
    }
    outIdx[row] = (float)bi;
  }
}

extern "C" void kernel_launch(void* const* d_in, const int* in_sizes, int n_in,
                              void* d_out, int out_size, void* d_ws, size_t ws_size,
                              hipStream_t stream) {
  const float* z = (const float*)d_in[0];   // [8192, 256]
  const float* E = (const float*)d_in[1];   // [4096, 256]
  float* out    = (float*)d_out;
  float* adj    = out;                                   // 8192*8192
  float* outIdx = out + (size_t)N_NODES * N_NODES;       // 8192 (indices as float)

  // workspace layout
  float* rn   = (float*)d_ws;                            // 4096 f32
  float* pval = rn + N_CODES;                            // NCHUNKS*8192 f32
  int*   pidx = (int*)(pval + (size_t)NCHUNKS * N_NODES);// NCHUNKS*8192 i32

  enorm_kernel<<<N_CODES / 256, 256, 0, stream>>>(E, rn);

  dim3 gA(N_NODES / 128, N_NODES / 256);
  adj_sigmoid_kernel<<<gA, dim3(256), 0, stream>>>(z, adj);

  dim3 gB(NCHUNKS, N_NODES / 128);
  cos_partial_kernel<<<gB, dim3(256), 0, stream>>>(z, E, rn, pval, pidx);

  argmax_finalize_kernel<<<N_NODES / 256, 256, 0, stream>>>(pval, pidx, outIdx);
}